// Model_39676907884675
// MI455X (gfx1250) — compile-verified
//
#include <hip/hip_runtime.h>
#include <hip/hip_bf16.h>

// Fused SPD pipeline for MI455X (gfx1250, wave32):
//   v1 = conv7x7(x1,w)+b
//   L  = chol(v1)           (blocked NB=16; trailing SYRK on V_WMMA_F32_16X16X4_F32,
//                            C-tile preloaded as accumulator, A negated -> pure-store epilogue)
//   V  = inv(L)             (blocked: 16x16 diag inverses + per-block-column WMMA TRSM)
//   A  = V^T V = inv(v1)    (symmetric WMMA tiles, K unrolled x4)
//   out[i][j] = (A[i][j]/L[i][i] + 1)^2 * v1[i][j]   (computed in place into LDS)
//   final store: one TENSOR_STORE_FROM_LDS DMA of the contiguous 112x112 result.
// One workgroup per matrix; everything stays in LDS (163,776 B <= 160 KB -> 2 WG/WGP).

#define Bsz   64
#define Cch   8
#define Nn    112
#define Kk    7
#define LD    116      // W/V row stride: 116*m mod 64 permutes -> row-strided reads conflict-free
#define LDM   112      // Mlds stride (lane-consecutive accesses only; also contiguous for TDM)
#define SS    18       // scratch stride: 18*m mod 64 distinct for m=0..15 -> transpose reads OK
#define TPB   256      // 8 wave32s
#define NWAVE 8
#define NB    16       // panel / WMMA tile size
#define NT    7        // 112/16 tiles per dim

typedef __attribute__((ext_vector_type(2))) float        v2f;
typedef __attribute__((ext_vector_type(8))) float        v8f;
typedef __attribute__((ext_vector_type(4))) unsigned int v4u;
typedef __attribute__((ext_vector_type(4))) int          v4i;
typedef __attribute__((ext_vector_type(8))) int          v8i;

__device__ static inline int rfl(int x) {
#if defined(__gfx1250__)
    return __builtin_amdgcn_readfirstlane(x);   // force SGPR -> scalar branches/loops
#else
    return x;                                    // host-pass placeholder
#endif
}

__device__ static inline v8f wmma4(v2f a, v2f b, v8f c) {
#if defined(__gfx1250__)
    return __builtin_amdgcn_wmma_f32_16x16x4_f32(
        false, a, false, b, (short)0, c, false, false);
#else
    c[0] += a.x * b.x;                           // host-pass placeholder
    return c;
#endif
}

__global__ __launch_bounds__(TPB)
void spd_inv_fused_kernel(const float* __restrict__ x1,
                          const float* __restrict__ weight,
                          const float* __restrict__ bias,
                          float* __restrict__ out)
{
    extern __shared__ float smem[];
    float* Mlds = smem;                    // v1 -> final result (contiguous 112x112)
    float* W    = smem + Nn * LDM;         // chol workspace -> L, stride LD
    float* V    = W + Nn * LD;             // inv(L) (lower; upper zeroed), stride LD
    float* dinv = V + Nn * LD;             // 1 / L[i][i]  (112)
    float* xtra = dinv + Nn;               // union: conv weights (392) | per-wave scratch (8*16*SS)

    const int tid   = threadIdx.x;
    const int wave  = tid >> 5;
    const int lane  = tid & 31;
    const int laneM = lane & 15;
    const int kh    = lane >> 4;           // K-half this lane feeds (K=0,1 vs K=2,3)

    const int mat = blockIdx.x;            // 0..511 == (b, c_out)
    const int b   = mat / Cch;
    const int co  = mat % Cch;
    const float* xin = x1 + (size_t)b * Cch * Nn * Nn;
    float* po = out + (size_t)mat * Nn * Nn;

    // Stage conv weights for this output channel into LDS (xtra reused later as scratch).
    for (int i = tid; i < Cch * Kk * Kk; i += TPB)
        xtra[i] = weight[co * Cch * Kk * Kk + i];
    const float bv = bias[co];

    // gfx1250 global_prefetch_b8: warm caches with the 8 input planes.
    for (int i = tid; i < Cch * Nn; i += TPB)
        __builtin_prefetch(xin + (size_t)i * Nn, 0, 0);

    __syncthreads();

    // ---- Phase 1: v1 = Conv2d(8->8, 7x7, pad 3) + bias ------------------
    for (int p = tid; p < Nn * Nn; p += TPB) {
        const int i = p / Nn, j = p % Nn;
        float s = bv;
        for (int ci = 0; ci < Cch; ++ci) {
            const float* xc = xin + ci * Nn * Nn;
            const float* wc = xtra + ci * Kk * Kk;
            for (int dy = 0; dy < Kk; ++dy) {
                const int y = i + dy - 3;
                if ((unsigned)y >= (unsigned)Nn) continue;
                const float* xr = xc + y * Nn;
                for (int dx = 0; dx < Kk; ++dx) {
                    const int x = j + dx - 3;
                    if ((unsigned)x < (unsigned)Nn)
                        s = fmaf(xr[x], wc[dy * Kk + dx], s);
                }
            }
        }
        Mlds[i * LDM + j] = s;
        W[i * LD + j]     = s;
    }
    for (int p = tid; p < Nn * LD; p += TPB) V[p] = 0.0f;  // zero inv(L) buffer
    __syncthreads();

    // ---- Phase 2: blocked Cholesky (NB=16), SYRK trailing update on WMMA -
    for (int kb = 0; kb < NT; ++kb) {
        const int kp   = kb << 4;
        const int pend = kp + NB;

        // 2a: unblocked panel factorization. Each thread owns one trailing row
        //     (rows <= 111 < 256) -> no div/mod, l_ik hoisted.
        for (int k = kp; k < pend; ++k) {
            const float dk  = sqrtf(W[k * LD + k]);   // stable read (written last step)
            const float rdk = 1.0f / dk;
            const int   i   = k + 1 + tid;
            float lik = 0.0f;
            if (i < Nn) {
                lik = W[i * LD + k] * rdk;
                W[i * LD + k] = lik;
            }
            __syncthreads();
            if (i < Nn) {
                for (int j = k + 1; j < pend; ++j)
                    W[i * LD + j] = fmaf(-lik, W[j * LD + k], W[i * LD + j]);
            }
            if (tid == 0) W[k * LD + k] = dk;         // no reader of (k,k) in this pass
            __syncthreads();
        }

        // 2b: trailing SYRK  S22 -= L21*L21^T : C preloaded as accumulator, A negated.
        const int nb = NT - (kb + 1);
        if (nb > 0) {
            const int np = nb * (nb + 1) / 2;
            for (int p = wave; p < np; p += NWAVE) {
                const int pu = rfl(p);
                int bc = 0, pp = pu;
                while (pp >= nb - bc) { pp -= nb - bc; ++bc; }
                const int br = bc + pp;               // br >= bc
                const int i0 = (kb + 1 + br) << 4;
                const int j0 = (kb + 1 + bc) << 4;
                const int ri = (i0 + laneM) * LD;
                const int rj = (j0 + laneM) * LD;
                const int ka = kp + (kh << 1);
                v8f acc;
                #pragma unroll
                for (int r = 0; r < 8; ++r)           // C tile -> accumulator
                    acc[r] = W[(i0 + r + (kh << 3)) * LD + j0 + laneM];
                v2f a0, a1, a2, a3, b0, b1, b2, b3;   // A negated: D = C - L21*L21^T
                a0.x = -W[ri + ka +  0]; a0.y = -W[ri + ka +  1];
                a1.x = -W[ri + ka +  4]; a1.y = -W[ri + ka +  5];
                a2.x = -W[ri + ka +  8]; a2.y = -W[ri + ka +  9];
                a3.x = -W[ri + ka + 12]; a3.y = -W[ri + ka + 13];
                b0.x =  W[rj + ka +  0]; b0.y =  W[rj + ka +  1];
                b1.x =  W[rj + ka +  4]; b1.y =  W[rj + ka +  5];
                b2.x =  W[rj + ka +  8]; b2.y =  W[rj + ka +  9];
                b3.x =  W[rj + ka + 12]; b3.y =  W[rj + ka + 13];
                acc = wmma4(a0, b0, acc);
                acc = wmma4(a1, b1, acc);
                acc = wmma4(a2, b2, acc);
                acc = wmma4(a3, b3, acc);
                #pragma unroll
                for (int r = 0; r < 8; ++r)           // pure stores, no RMW
                    W[(i0 + r + (kh << 3)) * LD + j0 + laneM] = acc[r];
                if (br != bc) {                       // uniform scalar branch
                    #pragma unroll
                    for (int r = 0; r < 8; ++r)
                        W[(j0 + laneM) * LD + i0 + r + (kh << 3)] = acc[r];
                }
            }
        }
        __syncthreads();
    }

    // ---- Phase 3a: invert the seven 16x16 diagonal blocks (+ dinv table) -
    if (tid < Nn) {
        dinv[tid] = 1.0f / W[tid * LD + tid];
        const int bb = tid >> 4;
        const int j  = tid & 15;
        const int o  = bb << 4;
        for (int i = j; i < NB; ++i) {
            float s = (i == j) ? 1.0f : 0.0f;
            for (int kk = j; kk < i; ++kk)
                s = fmaf(-W[(o + i) * LD + o + kk], V[(o + kk) * LD + o + j], s);
            V[(o + i) * LD + o + j] = s / W[(o + i) * LD + o + i];
        }
    }
    __syncthreads();

    // ---- Phase 3b: blocked TRSM on WMMA, one wave per block-column ------
    // X[bi][bj] = -D_bi * ( sum_{k=bj}^{bi-1} L[bi][k] * X[k][bj] )
    if (rfl(wave) < NT) {
        const int bj  = rfl(wave);
        const int cj0 = bj << 4;
        float* S = xtra + wave * (NB * SS);           // per-wave 16x16 T staging
        for (int bi = bj + 1; bi < NT; ++bi) {
            v8f t = {0.f, 0.f, 0.f, 0.f, 0.f, 0.f, 0.f, 0.f};
            for (int k = bj; k < bi; ++k) {
                const int ar = ((bi << 4) + laneM) * LD + (k << 4) + (kh << 1);
                const int bc = (k << 4) + (kh << 1);
                v2f a0, a1, a2, a3, b0, b1, b2, b3;
                a0.x = W[ar +  0]; a0.y = W[ar +  1];
                a1.x = W[ar +  4]; a1.y = W[ar +  5];
                a2.x = W[ar +  8]; a2.y = W[ar +  9];
                a3.x = W[ar + 12]; a3.y = W[ar + 13];
                b0.x = V[(bc +  0) * LD + cj0 + laneM]; b0.y = V[(bc +  1) * LD + cj0 + laneM];
                b1.x = V[(bc +  4) * LD + cj0 + laneM]; b1.y = V[(bc +  5) * LD + cj0 + laneM];
                b2.x = V[(bc +  8) * LD + cj0 + laneM]; b2.y = V[(bc +  9) * LD + cj0 + laneM];
                b3.x = V[(bc + 12) * LD + cj0 + laneM]; b3.y = V[(bc + 13) * LD + cj0 + laneM];
                t = wmma4(a0, b0, t);
                t = wmma4(a1, b1, t);
                t = wmma4(a2, b2, t);
                t = wmma4(a3, b3, t);
            }
            #pragma unroll
            for (int r = 0; r < 8; ++r)               // stage T (row-major, stride SS)
                S[(r + (kh << 3)) * SS + laneM] = t[r];
            // X[bi][bj] = -D_bi * T  (A = D_bi negated, B = T from scratch)
            const int dr = ((bi << 4) + laneM) * LD + (bi << 4) + (kh << 1);
            v2f a0, a1, a2, a3, b0, b1, b2, b3;
            a0.x = -V[dr +  0]; a0.y = -V[dr +  1];
            a1.x = -V[dr +  4]; a1.y = -V[dr +  5];
            a2.x = -V[dr +  8]; a2.y = -V[dr +  9];
            a3.x = -V[dr + 12]; a3.y = -V[dr + 13];
            b0.x = S[((kh << 1) +  0) * SS + laneM]; b0.y = S[((kh << 1) +  1) * SS + laneM];
            b1.x = S[((kh << 1) +  4) * SS + laneM]; b1.y = S[((kh << 1) +  5) * SS + laneM];
            b2.x = S[((kh << 1) +  8) * SS + laneM]; b2.y = S[((kh << 1) +  9) * SS + laneM];
            b3.x = S[((kh << 1) + 12) * SS + laneM]; b3.y = S[((kh << 1) + 13) * SS + laneM];
            v8f x = {0.f, 0.f, 0.f, 0.f, 0.f, 0.f, 0.f, 0.f};
            x = wmma4(a0, b0, x);
            x = wmma4(a1, b1, x);
            x = wmma4(a2, b2, x);
            x = wmma4(a3, b3, x);
            #pragma unroll
            for (int r = 0; r < 8; ++r)
                V[((bi << 4) + r + (kh << 3)) * LD + cj0 + laneM] = x[r];
        }
    }
    __syncthreads();

    // ---- Phase 4: A = V^T * V, symmetric tiles; result written in place --
    // Each (bi<=bj) pair exclusively owns regions (bi,bj) and (bj,bi) of Mlds.
    float* S = xtra + wave * (NB * SS);               // per-wave transpose staging
    for (int p = wave; p < NT * (NT + 1) / 2; p += NWAVE) {
        const int pu = rfl(p);
        int bi = 0, pp = pu;                          // (bi <= bj) tile pair
        while (pp >= NT - bi) { pp -= NT - bi; ++bi; }
        const int bj = bi + pp;
        const int ci = (bi << 4) + laneM;
        const int cj = (bj << 4) + laneM;
        v8f acc = {0.f, 0.f, 0.f, 0.f, 0.f, 0.f, 0.f, 0.f};
        for (int kb = rfl(bj); kb < NT; ++kb) {       // V[k][c]==0 for k<c
            const int ka = (kb << 4) + (kh << 1);
            v2f a0, a1, a2, a3, b0, b1, b2, b3;
            a0.x = V[(ka +  0) * LD + ci]; a0.y = V[(ka +  1) * LD + ci];
            a1.x = V[(ka +  4) * LD + ci]; a1.y = V[(ka +  5) * LD + ci];
            a2.x = V[(ka +  8) * LD + ci]; a2.y = V[(ka +  9) * LD + ci];
            a3.x = V[(ka + 12) * LD + ci]; a3.y = V[(ka + 13) * LD + ci];
            b0.x = V[(ka +  0) * LD + cj]; b0.y = V[(ka +  1) * LD + cj];
            b1.x = V[(ka +  4) * LD + cj]; b1.y = V[(ka +  5) * LD + cj];
            b2.x = V[(ka +  8) * LD + cj]; b2.y = V[(ka +  9) * LD + cj];
            b3.x = V[(ka + 12) * LD + cj]; b3.y = V[(ka + 13) * LD + cj];
            acc = wmma4(a0, b0, acc);
            acc = wmma4(a1, b1, acc);
            acc = wmma4(a2, b2, acc);
            acc = wmma4(a3, b3, acc);
        }
        #pragma unroll
        for (int r = 0; r < 8; ++r)                   // stage tile for transposed reuse
            S[(r + (kh << 3)) * SS + laneM] = acc[r];
        // direct tile: out[i][j] = (A*dinv[i]+1)^2 * v1[i][j], in place
        #pragma unroll
        for (int r = 0; r < 8; ++r) {
            const int ii = (bi << 4) + r + (kh << 3);
            const int jj = (bj << 4) + laneM;
            const float mm = Mlds[ii * LDM + jj];
            const float s  = fmaf(acc[r], dinv[ii], 1.0f);
            Mlds[ii * LDM + jj] = s * s * mm;
        }
        if (bi != bj) {                               // mirror tile via transposed staging
            #pragma unroll
            for (int r = 0; r < 8; ++r) {
                const int row = (bj << 4) + r + (kh << 3);   // output row (j-side)
                const int col = (bi << 4) + laneM;           // output col (i-side)
                const float av  = S[laneM * SS + r + (kh << 3)];   // A[col][row]
                const float mmT = Mlds[row * LDM + col];           // v1 symmetric
                const float s2  = fmaf(av, dinv[row], 1.0f);       // dinv broadcast
                Mlds[row * LDM + col] = s2 * s2 * mmT;
            }
        }
    }
    __syncthreads();

    // ---- Phase 5: DMA the contiguous result to HBM via the Tensor Data Mover.
#if defined(__gfx1250__) && __has_builtin(__builtin_amdgcn_tensor_store_from_lds)
    if (rfl(wave) == 0) {
        const unsigned long long ga = (unsigned long long)(uintptr_t)po;
        const unsigned lds = (unsigned)(uintptr_t)Mlds;   // flat LDS addr low 32 = LDS offset
        v4u g0;
        g0[0] = 1u;                                        // count=1 (valid descriptor)
        g0[1] = lds;                                       // lds_addr
        g0[2] = (unsigned)(ga & 0xFFFFFFFFu);              // global_addr[31:0]
        g0[3] = (unsigned)((ga >> 32) & 0x01FFFFFFu) | (2u << 30);  // addr[56:32] | type=2
        v8i g1;
        g1[0] = 0x20000;                                   // data_size = 4 bytes
        g1[1] = (int)((Nn * Nn) << 16);                    // tensor_dim0 = 12544 (lo16)
        g1[2] = 1 << 16;                                   // tensor_dim1 = 1
        g1[3] = (int)((Nn * Nn) << 16);                    // tile_dim0 = 12544
        g1[4] = 1;                                         // tile_dim1 = 1
        g1[5] = Nn * Nn;                                   // tensor_dim0_stride
        g1[6] = 0;
        g1[7] = 0;
        const v4i gz  = {0, 0, 0, 0};                      // groups 2/3 unused (<=2D)
        const v8i gz8 = {0, 0, 0, 0, 0, 0, 0, 0};          // 6-arg toolchain extra group
        __builtin_amdgcn_tensor_store_from_lds(g0, g1, gz, gz, gz8, 0);
        __builtin_amdgcn_s_wait_tensorcnt(0);              // TDM reads LDS: hold the workgroup
    }
    __syncthreads();
#else
    for (int p = tid; p < Nn * Nn; p += TPB) po[p] = Mlds[p];  // fallback copy
#endif
}

extern "C" void kernel_launch(void* const* d_in, const int* in_sizes, int n_in,
                              void* d_out, int out_size, void* d_ws, size_t ws_size,
                              hipStream_t stream) {
    const float* x1 = (const float*)d_in[0];  // (64, 8, 112, 112) fp32
    const float* w  = (const float*)d_in[1];  // (8, 8, 7, 7) fp32
    const float* bs = (const float*)d_in[2];  // (8,) fp32
    float* out = (float*)d_out;               // (64, 8, 112, 112) fp32

    // Mlds (112x112) + W,V (112x116) + dinv (112) + union(weights, 8-wave scratch).
    const int scrN  = NWAVE * NB * SS;
    const int xtraN = (Cch * Kk * Kk > scrN) ? Cch * Kk * Kk : scrN;
    const size_t shmem =
        (size_t)(Nn * LDM + 2 * Nn * LD + Nn + xtraN) * sizeof(float);
    spd_inv_fused_kernel<<<Bsz * Cch, TPB, shmem, stream>>>(x1, w, bs, out);
}